// Conv_Layer_Activation_27221502722284
// MI455X (gfx1250) — compile-verified
//
#include <hip/hip_runtime.h>
#include <hip/hip_bf16.h>
#include <math.h>

// Elementwise activation-select over x[B, N], codes[N] in 0..5, fp32.
// ~207 MB HBM traffic => ~8.9 us floor at 23.3 TB/s. Previous version was
// issue-bound (~160 instr / KB / wave). This version: each thread owns 4
// columns x 8 batch rows, hoisting the 20 code-compare lane masks (SGPRs)
// and address setup out of the stream loop => ~100 instr / KB / wave.

typedef __attribute__((ext_vector_type(4))) float fv4;
typedef __attribute__((ext_vector_type(4))) int   iv4;

#ifndef ROWS_PER_THREAD
#define ROWS_PER_THREAD 8
#endif

__device__ __forceinline__ float fast_tanh(float x) {
#if __has_builtin(__builtin_amdgcn_tanhf)
  return __builtin_amdgcn_tanhf(x);      // gfx1250 V_TANH_F32 (trans pipe)
#else
  return tanhf(x);
#endif
}

// Loop-invariant per-column selection predicates: the compiler keeps these as
// wave-wide lane masks in SGPRs, so the inner loop pays only v_cndmask.
struct Sel { bool s1, s2, s3, s4, s5; };

__device__ __forceinline__ Sel make_sel(int code) {
  Sel s;
  s.s1 = (code == 1);   // sigmoid
  s.s2 = (code == 2);   // tanh
  s.s3 = (code == 3);   // elu
  s.s4 = (code == 4);   // leaky_relu(0.01)
  s.s5 = (code == 5);   // gelu (tanh approx)
  return s;
}

__device__ __forceinline__ float apply_act(float x, const Sel& s) {
  // relu: 1 op
  float r  = fmaxf(x, 0.0f);
  // leaky: exact via max(x, 0.01x) for both signs: 2 ops
  float lk = fmaxf(x, 0.01f * x);
  // tanh: hw trans: 1 op
  float th = fast_tanh(x);
  // sigmoid = 0.5*tanh(x/2) + 0.5 (exact identity, tail-safe): 3 ops
  float sig = fmaf(0.5f, fast_tanh(0.5f * x), 0.5f);
  // elu = x>0 ? x : exp(x)-1 (selected path never overflows): 5 ops
  float ex = __expf(x);
  float el = (x > 0.0f) ? x : (ex - 1.0f);
  // gelu tanh approx: 0.5x(1+tanh(sqrt(2/pi)(x+0.044715x^3))): ~6 ops
  const float k0 = 0.7978845608028654f;              // sqrt(2/pi)
  const float k1 = 0.044715f * 0.7978845608028654f;
  float x2 = x * x;
  float u  = x * fmaf(k1, x2, k0);
  float hx = 0.5f * x;
  float g  = fmaf(hx, fast_tanh(u), hx);
  // 5 cndmasks; cmps hoisted out of the B loop
  r = s.s1 ? sig : r;
  r = s.s2 ? th  : r;
  r = s.s3 ? el  : r;
  r = s.s4 ? lk  : r;
  r = s.s5 ? g   : r;
  return r;
}

__global__ void __launch_bounds__(256)
act_select_v4r(const fv4* __restrict__ x, const iv4* __restrict__ codes,
               fv4* __restrict__ out, int N4, int B) {
  int n4 = blockIdx.x * blockDim.x + threadIdx.x;
  if (n4 >= N4) return;

  // codes: hot + reused (8 in-thread rows, 8 grid.y chunks) -> temporal load
  iv4 c = codes[n4];
  Sel sx = make_sel(c.x), sy = make_sel(c.y), sz = make_sel(c.z), sw = make_sel(c.w);

  int b0 = blockIdx.y * ROWS_PER_THREAD;
  int rows = B - b0;
  if (rows > ROWS_PER_THREAD) rows = ROWS_PER_THREAD;

  size_t idx = (size_t)b0 * (size_t)N4 + (size_t)n4;
  const size_t stride = (size_t)N4;

#pragma unroll 2
  for (int r = 0; r < rows; ++r) {
    // Warm the line this thread touches in 2 iterations (speculative; OOB
    // translations silently dropped per gfx1250 prefetch semantics).
    __builtin_prefetch((const void*)(x + idx + 2 * stride), 0, 0);

    // x/out streamed exactly once -> non-temporal, so the ~205MB stream does
    // not evict the hot 1.6MB codes array from the 192MB L2.
    fv4 v = __builtin_nontemporal_load(x + idx);
    fv4 o;
    o.x = apply_act(v.x, sx);
    o.y = apply_act(v.y, sy);
    o.z = apply_act(v.z, sz);
    o.w = apply_act(v.w, sw);
    __builtin_nontemporal_store(o, out + idx);

    idx += stride;
  }
}

// Scalar tail for N % 4 != 0 (not hit for N = 128*56*56; kept for generality).
__global__ void __launch_bounds__(256)
act_select_tail(const float* __restrict__ x, const int* __restrict__ codes,
                float* __restrict__ out, int N, int start) {
  int n = start + blockIdx.x * blockDim.x + threadIdx.x;
  if (n >= N) return;
  Sel s = make_sel(codes[n]);
  size_t idx = (size_t)blockIdx.y * (size_t)N + (size_t)n;
  float v = __builtin_nontemporal_load(x + idx);
  __builtin_nontemporal_store(apply_act(v, s), out + idx);
}

extern "C" void kernel_launch(void* const* d_in, const int* in_sizes, int n_in,
                              void* d_out, int out_size, void* d_ws, size_t ws_size,
                              hipStream_t stream) {
  const float* x   = (const float*)d_in[0];
  const int* codes = (const int*)d_in[1];
  float* out       = (float*)d_out;

  const int total = in_sizes[0];       // B * N
  const int N     = in_sizes[1];       // C*H*W = 401408
  const int B     = total / N;         // 64
  const int N4    = N / 4;             // 100352
  const int tail  = N - N4 * 4;

  dim3 blk(256);
  if (N4 > 0) {
    dim3 grd((unsigned)((N4 + 255) / 256),
             (unsigned)((B + ROWS_PER_THREAD - 1) / ROWS_PER_THREAD));
    act_select_v4r<<<grd, blk, 0, stream>>>(
        (const fv4*)x, (const iv4*)codes, (fv4*)out, N4, B);
  }
  if (tail > 0) {
    dim3 grd((unsigned)((tail + 255) / 256), (unsigned)B);
    act_select_tail<<<grd, blk, 0, stream>>>(x, codes, out, N, N4 * 4);
  }
}